// TaylorExp_24970939859448
// MI455X (gfx1250) — compile-verified
//
#include <hip/hip_runtime.h>

// TaylorExp feature map: per row of 16 floats, emit 273 floats:
//   out[0]      = 1
//   out[1+i]    = x[i] * 0.5                       (d^{-1/4}, d=16)
//   out[17+i*16+j] = x[i]*x[j] * (1/(4*sqrt(2)))   (outer / sqrt(2) / sqrt(d))
//
// One wave32 per row. The 16x16 outer product is computed with a single
// V_WMMA_F32_16X16X4_F32: A's K=0 column = x, B's K=0 row = x, all other
// K slots zero, so D = x x^T.

typedef __attribute__((ext_vector_type(2))) float v2f;
typedef __attribute__((ext_vector_type(8))) float v8f;

__global__ void taylor_exp_wmma_kernel(const float* __restrict__ x,
                                       float* __restrict__ out,
                                       int nrows) {
    const int wave = (int)((blockIdx.x * blockDim.x + threadIdx.x) >> 5);
    const int lane = (int)(threadIdx.x & 31);
    if (wave >= nrows) return;  // wave-uniform branch: whole wave exits or stays

    const float* __restrict__ xr = x + (size_t)wave * 16;
    // Clamped address + select keeps all 32 lanes active (EXEC must be all-1s
    // for WMMA), while lanes 16-31 contribute zeros.
    float xl = xr[lane & 15];
    xl = (lane < 16) ? xl : 0.0f;

    // A 16x4 f32: VGPR0 lanes 0-15 = column K=0 (M=lane). a.y (K=1/K=3) = 0.
    // B 4x16 f32: VGPR0 lanes 0-15 = row K=0 (N=lane).    b.y = 0.
    v2f a;
    a.x = xl;
    a.y = 0.0f;
    v2f b = a;
    v8f c = {0.0f, 0.0f, 0.0f, 0.0f, 0.0f, 0.0f, 0.0f, 0.0f};

    // D = A x B + C  ->  D[m][n] = x[m]*x[n]
    c = __builtin_amdgcn_wmma_f32_16x16x4_f32(
        /*neg_a=*/false, a, /*neg_b=*/false, b,
        /*c_mod=*/(short)0, c, /*reuse_a=*/false, /*reuse_b=*/false);

    float* __restrict__ orow = out + (size_t)wave * 273;

    // Constant + linear terms (divergent, but after the WMMA).
    if (lane == 0) orow[0] = 1.0f;
    if (lane < 16) orow[1 + lane] = xl * 0.5f;

    // Quadratic terms, scaled by 1/(sqrt(2)*sqrt(16)).
    const float s = 0.17677669529663687f;  // 1/(4*sqrt(2))
    const int m0 = (lane < 16) ? 0 : 8;
    const int n  = lane & 15;
#pragma unroll
    for (int r = 0; r < 8; ++r) {
        // VGPR r, lanes 0-15: D[r][lane]; lanes 16-31: D[r+8][lane-16]
        orow[17 + (m0 + r) * 16 + n] = c[r] * s;
    }
}

extern "C" void kernel_launch(void* const* d_in, const int* in_sizes, int n_in,
                              void* d_out, int out_size, void* d_ws, size_t ws_size,
                              hipStream_t stream) {
    const float* x = (const float*)d_in[0];
    float* out = (float*)d_out;
    const int nrows = in_sizes[0] / 16;  // 262144 rows of 16 floats

    const int waves_per_block = 8;                 // 256 threads = 8 wave32
    const int threads = waves_per_block * 32;
    const int blocks = (nrows + waves_per_block - 1) / waves_per_block;

    taylor_exp_wmma_kernel<<<blocks, threads, 0, stream>>>(x, out, nrows);
}